// PyG_SGC_32495722562261
// MI455X (gfx1250) — compile-verified
//
#include <hip/hip_runtime.h>

#define N_NODES 50000
#define N_EDGES 800000
#define IN_F 128
#define OUT_F 64

typedef __attribute__((ext_vector_type(2))) float v2f;
typedef __attribute__((ext_vector_type(8))) float v8f;

// ---------------- degree / normalization ----------------

__global__ void k_deg_init(float* __restrict__ deg) {
  int i = blockIdx.x * blockDim.x + threadIdx.x;
  if (i < N_NODES) deg[i] = 1.0f;  // self-loop contributes 1
}

__global__ void k_deg_edges(const int* __restrict__ ei, float* __restrict__ deg) {
  int e = blockIdx.x * blockDim.x + threadIdx.x;
  if (e < N_EDGES) atomicAdd(&deg[ei[N_EDGES + e]], 1.0f);  // dst row of edge_index
}

__global__ void k_deg_fin(float* __restrict__ deg) {
  int i = blockIdx.x * blockDim.x + threadIdx.x;
  if (i < N_NODES) deg[i] = rsqrtf(deg[i]);  // deg >= 1 always
}

// ---------------- propagation: out = Â @ in ----------------
// Self-loop term also fully initializes `out` (no memset needed):
// out[i,:] = dinv[i]^2 * in[i,:]
__global__ void k_hop_selfloop(const float4* __restrict__ in4,
                               const float* __restrict__ dinv,
                               float4* __restrict__ out4) {
  int t = blockIdx.x * blockDim.x + threadIdx.x;
  if (t >= N_NODES * 32) return;
  int node = t >> 5, q = t & 31;
  float w = dinv[node];
  w *= w;
  float4 v = in4[node * 32 + q];
  out4[node * 32 + q] = make_float4(w * v.x, w * v.y, w * v.z, w * v.w);
}

// One wave per edge; each lane moves a float4 (32 lanes * 16B = full 128-float row).
// Scatter via global_atomic_add_f32; working set is L2-resident (25.6MB << 192MB).
__global__ void k_hop_edges(const int* __restrict__ ei,
                            const float* __restrict__ dinv,
                            const float4* __restrict__ in4,
                            float* __restrict__ out) {
  int t = blockIdx.x * blockDim.x + threadIdx.x;
  if (t >= N_EDGES * 32) return;
  int e = t >> 5, q = t & 31;
  int s = ei[e];
  int d = ei[N_EDGES + e];
  float w = dinv[s] * dinv[d];
  float4 v = in4[s * 32 + q];
  float* o = out + (size_t)d * IN_F + q * 4;
  atomicAdd(o + 0, w * v.x);
  atomicAdd(o + 1, w * v.y);
  atomicAdd(o + 2, w * v.z);
  atomicAdd(o + 3, w * v.w);
}

// ---------------- final linear layer via V_WMMA_F32_16X16X4_F32 ----------------
// Y[50000,64] = X[50000,128] @ W.T + b.  One wave computes a 16x64 tile
// (4 accumulators of v8f). K=128 consumed 4 at a time -> 32*4 = 128 WMMA/wave.
__global__ __launch_bounds__(256) void k_gemm_wmma(const float* __restrict__ X,
                                                   const float* __restrict__ W,
                                                   const float* __restrict__ bias,
                                                   float* __restrict__ Y) {
  const int mtiles = N_NODES / 16;  // 3125 exactly (50000 % 16 == 0)
  int wave = blockIdx.x * (256 / 32) + (threadIdx.x >> 5);
  if (wave >= mtiles) return;  // wave-uniform guard: EXEC all-1s for WMMA
  int lane = threadIdx.x & 31;
  int half = lane >> 4;   // 0: K={kb,kb+1}, 1: K={kb+2,kb+3}
  int l = lane & 15;      // M row (A) / N col (B) within tile
  int m0 = wave * 16;

  // A: lane carries row (m0+l), two consecutive K values at offset half*2
  const float* ap = X + (size_t)(m0 + l) * IN_F + half * 2;
  // B[k][n] = W[n][k]; lane carries column n0+l
  const float* bp0 = W + (size_t)(l + 0)  * IN_F + half * 2;
  const float* bp1 = W + (size_t)(l + 16) * IN_F + half * 2;
  const float* bp2 = W + (size_t)(l + 32) * IN_F + half * 2;
  const float* bp3 = W + (size_t)(l + 48) * IN_F + half * 2;

  v8f acc0 = {}, acc1 = {}, acc2 = {}, acc3 = {};

#pragma unroll 8
  for (int kb = 0; kb < IN_F; kb += 4) {
    v2f a  = {ap[kb],  ap[kb + 1]};
    v2f b0 = {bp0[kb], bp0[kb + 1]};
    v2f b1 = {bp1[kb], bp1[kb + 1]};
    v2f b2 = {bp2[kb], bp2[kb + 1]};
    v2f b3 = {bp3[kb], bp3[kb + 1]};
    acc0 = __builtin_amdgcn_wmma_f32_16x16x4_f32(false, a, false, b0, (short)0, acc0, false, false);
    acc1 = __builtin_amdgcn_wmma_f32_16x16x4_f32(false, a, false, b1, (short)0, acc1, false, false);
    acc2 = __builtin_amdgcn_wmma_f32_16x16x4_f32(false, a, false, b2, (short)0, acc2, false, false);
    acc3 = __builtin_amdgcn_wmma_f32_16x16x4_f32(false, a, false, b3, (short)0, acc3, false, false);
  }

  // C/D layout: VGPR j -> M = j + half*8, N = l
  float bb0 = bias[l], bb1 = bias[16 + l], bb2 = bias[32 + l], bb3 = bias[48 + l];
  int rbase = m0 + half * 8;
#pragma unroll
  for (int j = 0; j < 8; ++j) {
    float* yr = Y + (size_t)(rbase + j) * OUT_F + l;
    yr[0]  = acc0[j] + bb0;
    yr[16] = acc1[j] + bb1;
    yr[32] = acc2[j] + bb2;
    yr[48] = acc3[j] + bb3;
  }
}

// ---------------- host orchestration ----------------

extern "C" void kernel_launch(void* const* d_in, const int* in_sizes, int n_in,
                              void* d_out, int out_size, void* d_ws, size_t ws_size,
                              hipStream_t stream) {
  (void)in_sizes; (void)n_in; (void)out_size; (void)ws_size;
  const float* x  = (const float*)d_in[0];   // [50000,128]
  const int*   ei = (const int*)d_in[1];     // [2,800000] flat: src then dst
  const float* W  = (const float*)d_in[2];   // [64,128]
  const float* b  = (const float*)d_in[3];   // [64]
  float* Y = (float*)d_out;                  // [50000,64]

  float* dinv = (float*)d_ws;                         // 50000 floats
  float* buf0 = dinv + 50048;                         // padded to 16-float mult
  float* buf1 = buf0 + (size_t)N_NODES * IN_F;        // 6.4M floats each

  const int T = 256;
  k_deg_init <<<(N_NODES + T - 1) / T, T, 0, stream>>>(dinv);
  k_deg_edges<<<(N_EDGES + T - 1) / T, T, 0, stream>>>(ei, dinv);
  k_deg_fin  <<<(N_NODES + T - 1) / T, T, 0, stream>>>(dinv);

  // hop 1: x -> buf0
  k_hop_selfloop<<<(N_NODES * 32 + T - 1) / T, T, 0, stream>>>((const float4*)x, dinv, (float4*)buf0);
  k_hop_edges   <<<(N_EDGES * 32 + T - 1) / T, T, 0, stream>>>(ei, dinv, (const float4*)x, buf0);
  // hop 2: buf0 -> buf1
  k_hop_selfloop<<<(N_NODES * 32 + T - 1) / T, T, 0, stream>>>((const float4*)buf0, dinv, (float4*)buf1);
  k_hop_edges   <<<(N_EDGES * 32 + T - 1) / T, T, 0, stream>>>(ei, dinv, (const float4*)buf0, buf1);

  // final linear layer: 3125 m-tiles, 8 waves/block
  k_gemm_wmma<<<(3125 + 7) / 8, T, 0, stream>>>(buf1, W, b, Y);
}